// Model_39840116637815
// MI455X (gfx1250) — compile-verified
//
#include <hip/hip_runtime.h>

// ---------------- problem constants (match reference) ----------------
#define BATCH 4
#define NV    5023
#define NBR   10
#define KKER  9
#define BV    (BATCH * NV)          // 20092 rows

typedef __attribute__((ext_vector_type(2))) float v2f;
typedef __attribute__((ext_vector_type(8))) float v8f;

// 1/sqrt(1 + 1e-3)  (BatchNorm inference, moving_var = 1)
#define BN_SCALE 0.9995003746877732f

// =====================================================================
// GEMM: C[M,N] = A[M,Kd] * B[Kd,N]  via V_WMMA_F32_16X16X4_F32
// One wave (32 lanes) per 16x16 output tile. N % 16 == 0, Kd % 4 == 0.
// A-matrix layout (16x4 f32): lanes 0-15 -> rows M=0..15 with K=kb..kb+1
// where kb = 0 (lanes<16) or 2 (lanes>=16). B(4x16): same K split, N=lane&15.
// C/D (16x16 f32, 8 VGPRs): row = r + 8*(lane>=16), col = lane&15.
// =====================================================================
__global__ __launch_bounds__(32) void gemm_wmma_f32(
    const float* __restrict__ A, const float* __restrict__ Bm,
    float* __restrict__ C, int M, int N, int Kd)
{
    const int tileN = blockIdx.x;
    const int tileM = blockIdx.y;
    const int lane  = threadIdx.x;
    const int half  = lane >> 4;          // 0 or 1
    const int l16   = lane & 15;

    const int arow    = tileM * 16 + l16;
    const bool avalid = (arow < M);
    const int arowc   = avalid ? arow : (M - 1);    // clamp to stay in bounds
    const int col     = tileN * 16 + l16;

    const float* __restrict__ Arow = A + (size_t)arowc * Kd;

    v8f acc = {};
    for (int kk = 0; kk < Kd; kk += 4) {
        const int kb = kk + half * 2;
        float a0 = Arow[kb];
        float a1 = Arow[kb + 1];
        v2f a, b;
        a.x = avalid ? a0 : 0.0f;
        a.y = avalid ? a1 : 0.0f;
        b.x = Bm[(size_t)kb * N + col];
        b.y = Bm[(size_t)(kb + 1) * N + col];
        acc = __builtin_amdgcn_wmma_f32_16x16x4_f32(
            /*neg_a=*/false, a, /*neg_b=*/false, b,
            /*c_mod=*/(short)0, acc, /*reuse_a=*/false, /*reuse_b=*/false);
    }

    const int rowbase = tileM * 16 + half * 8;
#pragma unroll
    for (int r = 0; r < 8; ++r) {
        const int row = rowbase + r;
        if (row < M) C[(size_t)row * N + col] = acc[r];
    }
}

// =====================================================================
// ux = X[BV,Fin] * U[Fin,K]   (K = 9, tiny GEMV per row)
// =====================================================================
__global__ void ux_kernel(const float* __restrict__ X, const float* __restrict__ U,
                          float* __restrict__ UX, int Fin)
{
    const int r = blockIdx.x * blockDim.x + threadIdx.x;
    if (r >= BV) return;
    const float* __restrict__ x = X + (size_t)r * Fin;
    float acc[KKER];
#pragma unroll
    for (int k = 0; k < KKER; ++k) acc[k] = 0.0f;
    for (int f = 0; f < Fin; ++f) {
        const float xv = x[f];
#pragma unroll
        for (int k = 0; k < KKER; ++k)
            acc[k] = fmaf(xv, U[f * KKER + k], acc[k]);
    }
    float* __restrict__ o = UX + (size_t)r * KKER;
#pragma unroll
    for (int k = 0; k < KKER; ++k) o[k] = acc[k];
}

// =====================================================================
// Attention aggregation + bias + BatchNorm(inference) + ReLU.
// One WAVE32 per (b,v); lanes spread over output channels so the
// wx-row gather is a contiguous, fully coalesced load per k.
// Softmax over K=9 is recomputed per lane (broadcast loads, TRANS ops).
// adj value 0 = padding (skip); value a -> row a-1.
// =====================================================================
template <int COUT>
__global__ __launch_bounds__(256) void aggregate_wave(
    const float* __restrict__ WX, const float* __restrict__ UX,
    const int*   __restrict__ adj,
    const float* __restrict__ bias, const float* __restrict__ cvec,
    const float* __restrict__ gamma, const float* __restrict__ beta,
    float* __restrict__ H)
{
    constexpr int CPT = (COUT + 31) / 32;   // channels per lane
    constexpr int KC  = KKER * COUT;

    const int wid  = (int)((blockIdx.x * blockDim.x + threadIdx.x) >> 5);
    if (wid >= BV) return;                  // wave-uniform exit
    const int lane = (int)(threadIdx.x & 31);
    const int b = wid / NV;
    const int v = wid - b * NV;

    // adjacency row (uniform across the wave) + prefetch neighbor wx rows
    int an[NBR];
    int deg = 0;
#pragma unroll
    for (int n = 0; n < NBR; ++n) {
        an[n] = adj[v * NBR + n];
        deg += (an[n] != 0) ? 1 : 0;
    }
#pragma unroll
    for (int n = 0; n < NBR; ++n) {
        if (an[n] != 0) {
            const float* wxr = WX + (size_t)(b * NV + an[n] - 1) * KC;
            // each lane prefetches a different 64B chunk of the row
            __builtin_prefetch(wxr + lane * 16, 0, 1);
        }
    }

    // center attention logits with kernel bias folded in
    float uxc[KKER];
    {
        const float* __restrict__ p = UX + (size_t)wid * KKER;
#pragma unroll
        for (int k = 0; k < KKER; ++k) uxc[k] = p[k] + cvec[k];
    }

    float acc[CPT];
#pragma unroll
    for (int j = 0; j < CPT; ++j) acc[j] = 0.0f;

    for (int n = 0; n < NBR; ++n) {
        const int a = an[n];
        if (a == 0) continue;               // padding neighbor: zero patch
        const int nrow = b * NV + (a - 1);

        // softmax over kernels (same value in every lane)
        const float* __restrict__ uxn = UX + (size_t)nrow * KKER;
        float t[KKER];
        float m = -3.402823466e38f;
#pragma unroll
        for (int k = 0; k < KKER; ++k) {
            t[k] = uxc[k] - uxn[k];
            m = fmaxf(m, t[k]);
        }
        float s = 0.0f;
#pragma unroll
        for (int k = 0; k < KKER; ++k) {
            t[k] = __expf(t[k] - m);
            s += t[k];
        }
        const float inv = 1.0f / s;

        const float* __restrict__ wxr = WX + (size_t)nrow * KC;
#pragma unroll
        for (int k = 0; k < KKER; ++k) {
            const float qk = t[k] * inv;
#pragma unroll
            for (int j = 0; j < CPT; ++j) {
                const int c = lane + j * 32;
                if (c < COUT)               // compile-time resolvable for 32/64
                    acc[j] = fmaf(qk, wxr[k * COUT + c], acc[j]);
            }
        }
    }

    const float invdeg = (deg > 0) ? (1.0f / (float)deg) : 0.0f;
    float* __restrict__ h = H + (size_t)wid * COUT;
#pragma unroll
    for (int j = 0; j < CPT; ++j) {
        const int c = lane + j * 32;
        if (c < COUT) {
            float val = acc[j] * invdeg + bias[c];
            val = gamma[c] * val * BN_SCALE + beta[c];
            h[c] = fmaxf(val, 0.0f);
        }
    }
}

// =====================================================================
// Final head: out = H[BV,16] * Wf[16,3] + bf
// =====================================================================
__global__ void final_kernel(const float* __restrict__ H, const float* __restrict__ Wf,
                             const float* __restrict__ bf, float* __restrict__ Out)
{
    const int r = blockIdx.x * blockDim.x + threadIdx.x;
    if (r >= BV) return;
    const float* __restrict__ h = H + (size_t)r * 16;
    float o0 = bf[0], o1 = bf[1], o2 = bf[2];
#pragma unroll
    for (int f = 0; f < 16; ++f) {
        const float hv = h[f];
        o0 = fmaf(hv, Wf[f * 3 + 0], o0);
        o1 = fmaf(hv, Wf[f * 3 + 1], o1);
        o2 = fmaf(hv, Wf[f * 3 + 2], o2);
    }
    Out[(size_t)r * 3 + 0] = o0;
    Out[(size_t)r * 3 + 1] = o1;
    Out[(size_t)r * 3 + 2] = o2;
}

// =====================================================================
// Orchestration
// =====================================================================
extern "C" void kernel_launch(void* const* d_in, const int* in_sizes, int n_in,
                              void* d_out, int out_size, void* d_ws, size_t ws_size,
                              hipStream_t stream)
{
    (void)in_sizes; (void)n_in; (void)out_size; (void)ws_size;

    const float* x   = (const float*)d_in[0];
    const int*   adj = (const int*)  d_in[1];
    const float* W1  = (const float*)d_in[2];
    const float* b1  = (const float*)d_in[3];
    const float* u1  = (const float*)d_in[4];
    const float* c1  = (const float*)d_in[5];
    const float* g1  = (const float*)d_in[6];
    const float* be1 = (const float*)d_in[7];
    const float* W2  = (const float*)d_in[8];
    const float* b2  = (const float*)d_in[9];
    const float* u2  = (const float*)d_in[10];
    const float* c2  = (const float*)d_in[11];
    const float* g2  = (const float*)d_in[12];
    const float* be2 = (const float*)d_in[13];
    const float* W3  = (const float*)d_in[14];
    const float* b3  = (const float*)d_in[15];
    const float* u3  = (const float*)d_in[16];
    const float* c3  = (const float*)d_in[17];
    const float* g3  = (const float*)d_in[18];
    const float* be3 = (const float*)d_in[19];
    const float* Wf  = (const float*)d_in[20];
    const float* bf  = (const float*)d_in[21];
    float* out = (float*)d_out;

    // workspace carve-up; wx buffer (46 MB) fits in MI455X 192MB L2
    char* ws = (char*)d_ws;
    float* wx = (float*)(ws);                                    // BV*576
    float* ux = (float*)(ws + (size_t)BV * 576 * 4);             // BV*9
    float* hA = (float*)(ws + (size_t)BV * 576 * 4 + (size_t)BV * 9 * 4);   // BV*64
    float* hB = (float*)(ws + (size_t)BV * 576 * 4 + (size_t)BV * 9 * 4
                             + (size_t)BV * 64 * 4);             // BV*32

    const int mtiles = (BV + 15) / 16;       // 1256
    const dim3 wblk(32, 1, 1);
    const int vthreads = 256;
    const int vblocks = (BV + vthreads - 1) / vthreads;
    const int wavesPerBlock = 256 / 32;      // 8 vertices per block
    const int ablocks = (BV + wavesPerBlock - 1) / wavesPerBlock;

    // ---- layer 1: Fin=32, Cout=64, K*Cout=576 ----
    gemm_wmma_f32<<<dim3(576 / 16, mtiles), wblk, 0, stream>>>(x, W1, wx, BV, 576, 32);
    ux_kernel<<<vblocks, vthreads, 0, stream>>>(x, u1, ux, 32);
    aggregate_wave<64><<<ablocks, 256, 0, stream>>>(wx, ux, adj, b1, c1, g1, be1, hA);

    // ---- layer 2: Fin=64, Cout=32, K*Cout=288 ----
    gemm_wmma_f32<<<dim3(288 / 16, mtiles), wblk, 0, stream>>>(hA, W2, wx, BV, 288, 64);
    ux_kernel<<<vblocks, vthreads, 0, stream>>>(hA, u2, ux, 64);
    aggregate_wave<32><<<ablocks, 256, 0, stream>>>(wx, ux, adj, b2, c2, g2, be2, hB);

    // ---- layer 3: Fin=32, Cout=16, K*Cout=144 ----
    gemm_wmma_f32<<<dim3(144 / 16, mtiles), wblk, 0, stream>>>(hB, W3, wx, BV, 144, 32);
    ux_kernel<<<vblocks, vthreads, 0, stream>>>(hB, u3, ux, 32);
    aggregate_wave<16><<<ablocks, 256, 0, stream>>>(wx, ux, adj, b3, c3, g3, be3, hA);

    // ---- final dense head 16 -> 3 ----
    final_kernel<<<vblocks, vthreads, 0, stream>>>(hA, Wf, bf, out);
}